// CSPNAccelerateRef_38122129719886
// MI455X (gfx1250) — compile-verified
//
#include <hip/hip_runtime.h>
#include <stdint.h>

// Problem constants (match the JAX reference)
#define BSZ      16
#define KH       352
#define KW       1216
#define NTAP     9
#define TILE     512                      // pixels per block tile
#define NTHREADS 128                      // 4 waves; 4 pixels/thread
#define LROW     (TILE + 8)               // staged row: 4-float halo each side -> 520 floats
#define NCHUNK   (LROW / 4)               // 130 float4 chunks per staged row
#define TILES_W  ((KW + TILE - 1) / TILE) // 3

typedef float f4  __attribute__((ext_vector_type(4)));
typedef int   vi4 __attribute__((vector_size(16)));

#if __has_builtin(__builtin_amdgcn_global_load_async_to_lds_b128)
#define HAVE_ASYNC_LDS 1
#else
#define HAVE_ASYNC_LDS 0
#endif

__device__ __forceinline__ void stage_chunk(const float* gp, float* lp) {
#if HAVE_ASYNC_LDS
    // async DMA 16B: global -> LDS, tracked by ASYNCcnt
    __builtin_amdgcn_global_load_async_to_lds_b128(
        (__attribute__((address_space(1))) vi4*)(uintptr_t)gp,
        (__attribute__((address_space(3))) vi4*)(uint32_t)(uintptr_t)lp,
        0, 0);
#else
    *(f4*)lp = *(const f4*)gp;
#endif
}

__device__ __forceinline__ void wait_async0() {
#if HAVE_ASYNC_LDS
#if __has_builtin(__builtin_amdgcn_s_wait_asynccnt)
    __builtin_amdgcn_s_wait_asynccnt(0);
#else
    asm volatile("s_wait_asynccnt 0" ::: "memory");
#endif
#endif
}

__global__ __launch_bounds__(NTHREADS)
void cspn_accel_kernel(const float* __restrict__ ker,
                       const float* __restrict__ inp,
                       const float* __restrict__ inp0,
                       float* __restrict__ out)
{
    __shared__ __align__(16) float smem[3 * LROW];

    const int tid  = threadIdx.x;
    int bi         = blockIdx.x;
    const int tile = bi % TILES_W;  bi /= TILES_W;
    const int h    = bi % KH;
    const int b    = bi / KH;
    const int x0   = tile * TILE;

    // 1) zero-fill staging buffer: provides zero-padding + OOB-tile zeros for free
    for (int i = tid; i < 3 * LROW; i += NTHREADS) smem[i] = 0.0f;
    __syncthreads();

    // 2) async-stage input rows h-1, h, h+1 (with 4-float halos) into LDS
    const float* inb = inp + (size_t)b * KH * KW;
    for (int i = tid; i < 3 * NCHUNK; i += NTHREADS) {
        const int rr = i / NCHUNK;
        const int c4 = i - rr * NCHUNK;
        const int gr = h - 1 + rr;
        const int gc = x0 - 4 + c4 * 4;
        if ((unsigned)gr < (unsigned)KH && (unsigned)gc < (unsigned)KW) {
            stage_chunk(inb + (size_t)gr * KW + gc, &smem[rr * LROW + c4 * 4]);
        }
    }
    wait_async0();
    __syncthreads();

    // 3) compute 4 pixels per thread, streaming the 9 kernel planes non-temporally
    const int px = x0 + tid * 4;
    if (px < KW) {
        // 3x6 LDS neighborhood covering all 3x3 taps of the 4 pixels
        float s[3][6];
        const int lc0 = 4 + tid * 4 - 1;
#pragma unroll
        for (int r = 0; r < 3; ++r)
#pragma unroll
            for (int c = 0; c < 6; ++c)
                s[r][c] = smem[r * LROW + lc0 + c];

        const size_t plane  = (size_t)KH * KW;
        const size_t rowoff = (size_t)h * KW + px;
        const float* kb = ker + (size_t)b * NTAP * plane + rowoff;
        const f4 c0 = *(const f4*)(inp0 + (size_t)b * plane + rowoff);

        f4 acc = (f4){0.f, 0.f, 0.f, 0.f};
#pragma unroll
        for (int t = 0; t < NTAP; ++t) {
            const f4 kv = __builtin_nontemporal_load((const f4*)(kb + (size_t)t * plane));
            if (t == 4) {
                acc += kv * c0;                      // center tap anchored to input0
            } else {
                const int dy = t / 3, dx = t % 3;
                const f4 v = (f4){ s[dy][dx + 0], s[dy][dx + 1],
                                   s[dy][dx + 2], s[dy][dx + 3] };
                acc += kv * v;
            }
        }
        __builtin_nontemporal_store(acc, (f4*)(out + (size_t)b * plane + rowoff));
    }
}

extern "C" void kernel_launch(void* const* d_in, const int* in_sizes, int n_in,
                              void* d_out, int out_size, void* d_ws, size_t ws_size,
                              hipStream_t stream) {
    (void)in_sizes; (void)n_in; (void)out_size; (void)d_ws; (void)ws_size;
    const float* ker  = (const float*)d_in[0];   // [16, 9, 352, 1216] f32
    const float* inp  = (const float*)d_in[1];   // [16, 1, 352, 1216] f32
    const float* inp0 = (const float*)d_in[2];   // [16, 1, 352, 1216] f32
    float*       out  = (float*)d_out;           // [16, 1, 352, 1216] f32

    dim3 grid(BSZ * KH * TILES_W);
    dim3 block(NTHREADS);
    cspn_accel_kernel<<<grid, block, 0, stream>>>(ker, inp, inp0, out);
}